// PositionEncoder_4063039062375
// MI455X (gfx1250) — compile-verified
//
#include <hip/hip_runtime.h>
#include <cstdint>

// ---------------------------------------------------------------------------
// PositionEncoder for MI455X (gfx1250).
// Output [B, H*W, 69] f32 : slot 0 = x[b,p], slots 1..10 = interleaved axis
// features (norm,sign,sin,cos,round(tan)) for x/y, slots 11..39 one-hot x,
// slots 40..68 one-hot y.  579 MB of stores -> pure bandwidth kernel.
// Roofline: ~25 us at 23.3 TB/s; design keeps VALU/byte far below balance.
// ---------------------------------------------------------------------------

typedef float v4f __attribute__((ext_vector_type(4)));

#define LDS_CAP 10240          // floats: supports (W+H) <= 2048 coordinates
#define PI_F 3.14159265358979323846f

#if defined(__has_builtin)
#  if __has_builtin(__builtin_amdgcn_global_load_async_to_lds_b32)
#    define HAVE_ASYNC_LDS 1
#  endif
#  if __has_builtin(__builtin_amdgcn_s_wait_asynccnt)
#    define HAVE_WAIT_ASYNC 1
#  endif
#endif

// ---------------------------------------------------------------------------
// Kernel A: build per-coordinate axis feature tables into d_ws.
// Layout: tab[x*5 + j] for x-axis (x in [0,W)), then tab[W*5 + y*5 + j].
// ---------------------------------------------------------------------------
__global__ void pe_build_tables(const int* __restrict__ hptr,
                                const int* __restrict__ wptr,
                                float* __restrict__ tab) {
    const int H = *hptr;
    const int W = *wptr;
    const int total = W + H;
    for (int i = blockIdx.x * blockDim.x + threadIdx.x; i < total;
         i += gridDim.x * blockDim.x) {
        const bool isX = (i < W);
        const int c    = isX ? i : (i - W);
        const int size = isX ? W : H;
        const float cf = (float)c;

        float norm, angle;
        if (size > 1) {
            const float den = (float)(size - 1);
            const float r   = cf / den;
            norm  = 2.0f * r - 1.0f;
            angle = PI_F * r;
        } else {
            norm  = 0.0f;
            angle = 0.0f;
        }
        const float d  = cf - 0.5f * (float)size;
        const float sg = (d > 0.0f) ? 1.0f : ((d < 0.0f) ? -1.0f : 0.0f);
        const float sn = sinf(angle);
        const float cs = cosf(angle);
        const float rt = rintf(tanf(angle));   // round-half-even, matches jnp.round

        const int base = isX ? (c * 5) : (W * 5 + c * 5);
        tab[base + 0] = norm;
        tab[base + 1] = sg;
        tab[base + 2] = sn;
        tab[base + 3] = cs;
        tab[base + 4] = rt;
    }
}

// ---------------------------------------------------------------------------
// Value of flat output element e = q*69 + f, where q = b*P + p.
// TabPtr is either a shared-derived or a global-derived float* so that
// address-space inference can emit ds_load / global_load respectively.
// ---------------------------------------------------------------------------
template <typename TabPtr>
__device__ __forceinline__ float pe_value(unsigned q, unsigned f,
                                          unsigned xx, unsigned yy,
                                          const float* __restrict__ xin,
                                          TabPtr xt, TabPtr yt) {
    if (f == 0u) return __builtin_nontemporal_load(xin + q);
    const unsigned s = f - 1u;
    if (s < 10u) {
        const unsigned j = s >> 1;
        return (s & 1u) ? yt[yy * 5u + j] : xt[xx * 5u + j];
    }
    if (s < 39u) return (xx == (s - 10u)) ? 1.0f : 0.0f;
    return (yy == (s - 39u)) ? 1.0f : 0.0f;
}

// ---------------------------------------------------------------------------
// Streaming loop, specialized on (pow2 geometry) x (table location).
// One float4 (16B) non-temporal store per thread-iteration.
// ---------------------------------------------------------------------------
template <bool PW2, typename TabPtr>
__device__ __forceinline__ void pe_stream(const float* __restrict__ xin,
                                          TabPtr xt, TabPtr yt,
                                          float* __restrict__ out,
                                          unsigned total,
                                          unsigned W, unsigned H, unsigned P,
                                          unsigned wlog) {
    const unsigned pmask   = P - 1u;
    const unsigned wmask   = W - 1u;
    const unsigned ngroups = total >> 2;
    const unsigned stride  = gridDim.x * blockDim.x;

    for (unsigned g = blockIdx.x * blockDim.x + threadIdx.x; g < ngroups;
         g += stride) {
        const unsigned e0 = g << 2;
        unsigned q = e0 / 69u;            // constant divisor -> magic multiply
        unsigned f = e0 - q * 69u;
        unsigned p, xx, yy;
        if (PW2) {
            p  = q & pmask;
            yy = p >> wlog;
            xx = p & wmask;
        } else {
            p  = q % P;
            yy = p / W;
            xx = p - yy * W;
        }

        v4f v;
#pragma unroll
        for (int k = 0; k < 4; ++k) {
            v[k] = pe_value(q, f, xx, yy, xin, xt, yt);
            if (++f == 69u) {             // carry into next position
                f = 0u;
                ++q;
                if (++xx == W) { xx = 0u; if (++yy == H) yy = 0u; }
                if (++p == P) p = 0u;
            }
        }
        __builtin_nontemporal_store(v, (v4f*)(out + e0));
    }

    // Tail (total not divisible by 4) — single thread, trivially small.
    if (blockIdx.x == 0 && threadIdx.x == 0) {
        for (unsigned e = ngroups << 2; e < total; ++e) {
            unsigned q = e / 69u;
            unsigned f = e - q * 69u;
            unsigned p, xx, yy;
            if (PW2) {
                p  = q & pmask;
                yy = p >> wlog;
                xx = p & wmask;
            } else {
                p  = q % P;
                yy = p / W;
                xx = p - yy * W;
            }
            out[e] = pe_value(q, f, xx, yy, xin, xt, yt);
        }
    }
}

// ---------------------------------------------------------------------------
// Kernel B: stage axis table into LDS via async global->LDS DMA, then stream.
// ---------------------------------------------------------------------------
__global__ void __launch_bounds__(256)
pe_encode(const float* __restrict__ xin,
          const int* __restrict__ hptr,
          const int* __restrict__ wptr,
          const float* __restrict__ tab,
          float* __restrict__ out,
          unsigned total) {
    __shared__ float sh[LDS_CAP];

    const unsigned W = (unsigned)*wptr;
    const unsigned H = (unsigned)*hptr;
    const unsigned P = W * H;
    const unsigned tcount  = (W + H) * 5u;
    const bool     use_lds = (tcount <= (unsigned)LDS_CAP);

    if (use_lds) {
#ifdef HAVE_ASYNC_LDS
        for (unsigned t = threadIdx.x; t < tcount; t += blockDim.x) {
            auto gsrc = (__attribute__((address_space(1))) int*)
                            (uintptr_t)(tab + t);
            auto ldst = (__attribute__((address_space(3))) int*)
                            (unsigned int)(uintptr_t)(sh + t);
            __builtin_amdgcn_global_load_async_to_lds_b32(gsrc, ldst, 0, 0);
        }
#  ifdef HAVE_WAIT_ASYNC
        __builtin_amdgcn_s_wait_asynccnt(0);
#  else
        asm volatile("s_wait_asynccnt 0" ::: "memory");
#  endif
#else
        for (unsigned t = threadIdx.x; t < tcount; t += blockDim.x)
            sh[t] = tab[t];
#endif
        __syncthreads();
    }

    const bool pw2 = ((W & (W - 1u)) == 0u) && ((H & (H - 1u)) == 0u) &&
                     (W != 0u) && (H != 0u);
    const unsigned wlog = pw2 ? (unsigned)__builtin_ctz(W) : 0u;

    if (use_lds) {
        // Table pointers derived *unconditionally* from the __shared__ array:
        // after inlining, address-space inference emits ds_load for lookups.
        const float* xt = &sh[0];
        const float* yt = &sh[W * 5u];
        if (pw2) pe_stream<true >(xin, xt, yt, out, total, W, H, P, wlog);
        else     pe_stream<false>(xin, xt, yt, out, total, W, H, P, wlog);
    } else {
        const float* xt = tab;
        const float* yt = tab + W * 5u;
        if (pw2) pe_stream<true >(xin, xt, yt, out, total, W, H, P, wlog);
        else     pe_stream<false>(xin, xt, yt, out, total, W, H, P, wlog);
    }
}

// ---------------------------------------------------------------------------
// Host launcher. d_in[0]=x (f32, B*H*W), d_in[1]=height (i32), d_in[2]=width.
// d_ws holds the (W+H)*5 float axis table (20 KB for 512x512).
// ---------------------------------------------------------------------------
extern "C" void kernel_launch(void* const* d_in, const int* in_sizes, int n_in,
                              void* d_out, int out_size, void* d_ws, size_t ws_size,
                              hipStream_t stream) {
    (void)in_sizes; (void)n_in; (void)ws_size;
    const float* x    = (const float*)d_in[0];
    const int*   hptr = (const int*)d_in[1];
    const int*   wptr = (const int*)d_in[2];
    float*       out  = (float*)d_out;
    float*       tab  = (float*)d_ws;

    // Kernel A: axis feature tables (grid-stride; sizes read on device).
    pe_build_tables<<<16, 256, 0, stream>>>(hptr, wptr, tab);

    // Kernel B: streaming encode.
    const unsigned total   = (unsigned)out_size;
    const unsigned ngroups = total >> 2;
    int threads = 256;
    long long want = ((long long)ngroups + threads - 1) / threads;
    int blocks = (int)(want < 1 ? 1 : (want > 4096 ? 4096 : want));
    pe_encode<<<blocks, threads, 0, stream>>>(x, hptr, wptr, tab, out, total);
}